// DGCNN_89086211654205
// MI455X (gfx1250) — compile-verified
//
#include <hip/hip_runtime.h>

#define S_    8
#define N_    50000
#define E_    1600000
#define FIN_  139
#define H_    32
#define K_    600
#define LAT_  65
#define FEAT_ 9537      /* 65 + 32*296 */
#define L1IN_ 19074
#define CAP_  4096
#define NBINS_ 8192

typedef __attribute__((ext_vector_type(2))) float v2f;
typedef __attribute__((ext_vector_type(8))) float v8f;

// ---------------------------------------------------------------- utilities
__global__ void fill_f32(float* __restrict__ p, float v, int n) {
  int i = blockIdx.x * blockDim.x + threadIdx.x;
  if (i < n) p[i] = v;
}
__global__ void fill_u32(unsigned* __restrict__ p, unsigned v, int n) {
  int i = blockIdx.x * blockDim.x + threadIdx.x;
  if (i < n) p[i] = v;
}

// ---------------------------------------------------------------- degree
__global__ void deg_accum(const int* __restrict__ dst, float* __restrict__ deg, int nE) {
  int i = blockIdx.x * blockDim.x + threadIdx.x;
  if (i < nE) atomicAdd(&deg[dst[i]], 1.0f);
}
__global__ void rsqrt_ip(float* __restrict__ d, int n) {
  int i = blockIdx.x * blockDim.x + threadIdx.x;
  if (i < n) d[i] = rsqrtf(d[i]);   // deg >= 1 always (self loop)
}

// ------------------------------------------------- WMMA f32 GEMM:  Y = X @ W
// X:[M x KD] row-major, W:[KD x 32] row-major, Y:[M x 32].
// 128 threads = 4 waves per block; each wave owns one 16-row tile.
// B is staged transposed+zero-padded in LDS (shared by block); each wave
// stages its own 16xKP X tile with coalesced row loads. All fragment reads
// are 8B-aligned LDS b64 loads; no divergent bounds checks.
template <int KD>
__global__ __launch_bounds__(128)
void gemm_wmma_nx32(const float* __restrict__ X, const float* __restrict__ W,
                    float* __restrict__ Y, int numTiles) {
  constexpr int KP = (KD + 3) & ~3;
  __shared__ float ldsB[32 * KP];        // [n][k], zero-padded rows k >= KD
  __shared__ float ldsX[4 * 16 * KP];    // per-wave 16xKP tiles

  const int tid  = threadIdx.x;
  const int wv   = tid >> 5;                       // wave in block: 0..3
  const int lane = tid & 31;
  const int half = lane >> 4;                      // 0: lanes 0-15, 1: 16-31
  const int l16  = lane & 15;
  const int tile = blockIdx.x * 4 + wv;

  // stage B transposed: ldsB[n*KP + k] = W[k*32 + n] (0 for k >= KD)
  for (int idx = tid; idx < 32 * KP; idx += 128) {
    int n = idx / KP, k = idx - n * KP;
    int kc = (k < KD) ? k : (KD - 1);              // branchless clamp
    float v = W[kc * 32 + n];
    ldsB[idx] = (k < KD) ? v : 0.f;
  }

  // stage this wave's X tile (coalesced along rows)
  if (tile < numTiles) {
    const float* xb = X + (size_t)tile * 16 * KD;
    float* lx = ldsX + wv * 16 * KP;
    for (int r = 0; r < 16; ++r) {
      const float* xr = xb + r * KD;
      for (int c = lane; c < KP; c += 32) {
        int cc = (c < KD) ? c : (KD - 1);          // in-bounds; pad value dead
        float v = xr[cc];
        lx[r * KP + c] = (c < KD) ? v : 0.f;
      }
    }
  }
  __syncthreads();

  if (tile >= numTiles) return;

  const float* la = ldsX + wv * 16 * KP + l16 * KP;   // A row for this lane
  const float* lb0 = ldsB + l16 * KP;                 // B col l16
  const float* lb1 = ldsB + (16 + l16) * KP;          // B col l16+16
  v8f c0 = {}; v8f c1 = {};
#pragma unroll 5
  for (int k0 = 0; k0 < KP; k0 += 4) {
    int ka = k0 + 2 * half;                           // even -> 8B aligned
    v2f a  = *(const v2f*)(la  + ka);
    v2f b0 = *(const v2f*)(lb0 + ka);
    v2f b1 = *(const v2f*)(lb1 + ka);
    c0 = __builtin_amdgcn_wmma_f32_16x16x4_f32(false, a, false, b0, (short)0, c0, false, false);
    c1 = __builtin_amdgcn_wmma_f32_16x16x4_f32(false, a, false, b1, (short)0, c1, false, false);
  }
  const int row0 = tile * 16;
#pragma unroll
  for (int v = 0; v < 8; ++v) {                       // D: VGPR v -> M=v (+8 hi)
    int m = v + 8 * half;
    Y[(size_t)(row0 + m) * 32 + l16]      = c0[v];
    Y[(size_t)(row0 + m) * 32 + 16 + l16] = c1[v];
  }
}

// ---------------------------------------------- edge scatter (32 channels)
__global__ __launch_bounds__(256)
void scatter32(const int* __restrict__ src, const int* __restrict__ dst,
               const float* __restrict__ hW, const float* __restrict__ dinv,
               float* __restrict__ agg, int nE) {
  int e = blockIdx.x * (blockDim.x >> 5) + (threadIdx.x >> 5);
  if (e >= nE) return;
  int lane = threadIdx.x & 31;
  int s = src[e], d = dst[e];
  float coef = dinv[s] * dinv[d];
  atomicAdd(&agg[(size_t)d * 32 + lane], hW[(size_t)s * 32 + lane] * coef);
}
__global__ void finalize32(const float* __restrict__ agg, const float* __restrict__ hW,
                           const float* __restrict__ dinv, const float* __restrict__ b,
                           float* __restrict__ out, int n) {
  int i = blockIdx.x * blockDim.x + threadIdx.x;
  if (i >= n) return;
  int node = i >> 5, c = i & 31;
  float di = dinv[node];
  out[i] = tanhf(agg[i] + hW[i] * di * di + b[c]);     // self-loop folded here
}

// ---------------------------------------------------------------- layer 3
__global__ void gemv32(const float* __restrict__ h2, const float* __restrict__ W2,
                       float* __restrict__ hW2, int n) {
  int i = blockIdx.x * blockDim.x + threadIdx.x;
  if (i >= n) return;
  const float* r = h2 + (size_t)i * 32;
  float s = 0.f;
#pragma unroll
  for (int c = 0; c < 32; ++c) s += r[c] * W2[c];
  hW2[i] = s;
}
__global__ void scatter1(const int* __restrict__ src, const int* __restrict__ dst,
                         const float* __restrict__ hW2, const float* __restrict__ dinv,
                         float* __restrict__ agg, int nE) {
  int e = blockIdx.x * blockDim.x + threadIdx.x;
  if (e >= nE) return;
  int s = src[e], d = dst[e];
  atomicAdd(&agg[d], hW2[s] * dinv[s] * dinv[d]);
}
__global__ void finalize1(const float* __restrict__ agg, const float* __restrict__ hW2,
                          const float* __restrict__ dinv, const float* __restrict__ b,
                          float* __restrict__ h3, int n) {
  int i = blockIdx.x * blockDim.x + threadIdx.x;
  if (i >= n) return;
  float di = dinv[i];
  h3[i] = tanhf(agg[i] + hW2[i] * di * di + b[0]);
}

// ------------------------------------------------------------- top-K select
__device__ __forceinline__ unsigned descKey(float v) {
  unsigned u = __float_as_uint(v);
  u ^= (u >> 31) ? 0xFFFFFFFFu : 0x80000000u;  // ascending-ordered
  return ~u;                                    // descending-ordered
}
__global__ void hist_kernel(const float* __restrict__ h3, unsigned* __restrict__ hist, int n) {
  int i = blockIdx.x * blockDim.x + threadIdx.x;
  if (i < n) atomicAdd(&hist[descKey(h3[i]) >> 19], 1u);
}
__global__ void find_thresh(const unsigned* __restrict__ hist, unsigned* __restrict__ meta) {
  if (threadIdx.x != 0) return;
  unsigned cum = 0, T = NBINS_ - 1;
  for (int b = 0; b < NBINS_; ++b) { cum += hist[b]; if (cum >= K_) { T = b; break; } }
  meta[1] = T;
  meta[0] = 0;   // reset compaction counter
}
__global__ void compact(const float* __restrict__ h3, unsigned* __restrict__ meta,
                        unsigned long long* __restrict__ cand, int n) {
  int i = blockIdx.x * blockDim.x + threadIdx.x;
  if (i >= n) return;
  unsigned dk = descKey(h3[i]);
  if ((dk >> 19) <= meta[1]) {
    unsigned pos = atomicAdd(&meta[0], 1u);
    if (pos < CAP_) cand[pos] = ((unsigned long long)dk << 32) | (unsigned)i;
  }
}
__global__ __launch_bounds__(1024)
void sort_topk(const unsigned long long* __restrict__ cand, const unsigned* __restrict__ meta,
               int* __restrict__ idx) {
  __shared__ unsigned long long sk[CAP_];
  unsigned cnt = meta[0]; if (cnt > CAP_) cnt = CAP_;
  for (int i = threadIdx.x; i < CAP_; i += 1024)
    sk[i] = (i < (int)cnt) ? cand[i] : 0xFFFFFFFFFFFFFFFFull;
  for (int k = 2; k <= CAP_; k <<= 1) {
    for (int j = k >> 1; j > 0; j >>= 1) {
      __syncthreads();
      for (int i = threadIdx.x; i < CAP_; i += 1024) {
        int ixj = i ^ j;
        if (ixj > i) {
          unsigned long long a = sk[i], b = sk[ixj];
          bool asc = ((i & k) == 0);
          if ((a > b) == asc) { sk[i] = b; sk[ixj] = a; }
        }
      }
    }
  }
  __syncthreads();
  for (int i = threadIdx.x; i < K_; i += 1024) idx[i] = (int)(sk[i] & 0xFFFFFFFFu);
}

// ------------------------------------------------------------- pool / convs
__global__ void gather_pool(const int* __restrict__ idx, const float* __restrict__ h1,
                            const float* __restrict__ h2, const float* __restrict__ h3,
                            float* __restrict__ pooled) {
  int id = blockIdx.x * blockDim.x + threadIdx.x;
  if (id >= K_ * LAT_) return;
  int j = id / LAT_, t = id % LAT_;
  int node = idx[j];
  float v = (t < 32) ? h1[(size_t)node * 32 + t]
          : (t < 64) ? h2[(size_t)node * 32 + (t - 32)]
                     : h3[node];
  pooled[id] = v;
}
__global__ void emb_kernel(const float* __restrict__ h1, const float* __restrict__ h2,
                           const float* __restrict__ h3, float* __restrict__ featsS) {
  int t = threadIdx.x;
  if (t >= LAT_) return;
  float a = (t < 32) ? h1[t] : (t < 64) ? h2[t - 32] : h3[0];
  float b = (t < 32) ? h1[32 + t] : (t < 64) ? h2[32 + (t - 32)] : h3[1];
  featsS[t] = a * b;
}
__global__ void conv1_kernel(const float* __restrict__ pooled, const float* __restrict__ w,
                             const float* __restrict__ b, float* __restrict__ z1) {
  int id = blockIdx.x * blockDim.x + threadIdx.x;
  if (id >= 16 * K_) return;
  int c = id / K_, k = id % K_;
  const float* row = pooled + (size_t)k * LAT_;
  const float* wc  = w + (size_t)c * LAT_;
  float s = b[c];
  for (int t = 0; t < LAT_; ++t) s += row[t] * wc[t];
  z1[(size_t)c * K_ + k] = fmaxf(s, 0.f);
}
__global__ void conv2_kernel(const float* __restrict__ z1, const float* __restrict__ w,
                             const float* __restrict__ b, float* __restrict__ featsS) {
  int id = blockIdx.x * blockDim.x + threadIdx.x;
  if (id >= 32 * 296) return;
  int c = id / 296, k = id % 296;
  float s = b[c];
  for (int i = 0; i < 16; ++i) {
    const float* zi = z1 + (size_t)i * K_;
    const float* wi = w + ((size_t)c * 16 + i) * 5;
#pragma unroll
    for (int t = 0; t < 5; ++t) {
      int x = k + t;
      s += fmaxf(zi[2 * x], zi[2 * x + 1]) * wi[t];
    }
  }
  featsS[65 + c * 296 + k] = fmaxf(s, 0.f);
}

// -------------------------------------------------------------- head layers
__global__ void combine_kernel(const float* __restrict__ feats, const float* __restrict__ wp,
                               float* __restrict__ inputs) {
  int t = blockIdx.x * blockDim.x + threadIdx.x;
  if (t >= L1IN_) return;
  if (t < FEAT_) {
    float omw = 1.f - wp[0];
    float acc = feats[6 * FEAT_ + t];
    float p = 1.f;
    for (int c = 6; c >= 1; --c) { acc += p * feats[(c - 1) * FEAT_ + t]; p *= omw; }
    inputs[t] = acc;
  } else {
    inputs[t] = feats[7 * FEAT_ + (t - FEAT_)];
  }
}
__global__ __launch_bounds__(256)
void lin1_kernel(const float* __restrict__ inputs, const float* __restrict__ W,
                 const float* __restrict__ b, float* __restrict__ out) {
  __shared__ float red[256];
  int o = blockIdx.x;                       // 0..127
  float s = 0.f;
  for (int t = threadIdx.x; t < L1IN_; t += 256) s += inputs[t] * W[(size_t)t * 128 + o];
  red[threadIdx.x] = s; __syncthreads();
  for (int st = 128; st > 0; st >>= 1) {
    if (threadIdx.x < st) red[threadIdx.x] += red[threadIdx.x + st];
    __syncthreads();
  }
  if (threadIdx.x == 0) out[o] = fmaxf(red[0] + b[o], 0.f);
}
__global__ void lin4_kernel(const float* __restrict__ h, const float* __restrict__ W,
                            const float* __restrict__ b, float* __restrict__ out) {
  __shared__ float red[128];
  int t = threadIdx.x;
  red[t] = h[t] * W[t]; __syncthreads();
  for (int st = 64; st > 0; st >>= 1) {
    if (t < st) red[t] += red[t + st];
    __syncthreads();
  }
  if (t == 0) out[0] = red[0] + b[0];
}

// =========================================================== host launcher
extern "C" void kernel_launch(void* const* d_in, const int* in_sizes, int n_in,
                              void* d_out, int out_size, void* d_ws, size_t ws_size,
                              hipStream_t stream) {
  (void)in_sizes; (void)n_in; (void)out_size; (void)ws_size;
  const float* x   = (const float*)d_in[0];
  const int*   ei  = (const int*)d_in[1];     // [S,2,E]
  const float* W0  = (const float*)d_in[3];
  const float* b0  = (const float*)d_in[4];
  const float* W1  = (const float*)d_in[5];
  const float* b1  = (const float*)d_in[6];
  const float* W2  = (const float*)d_in[7];
  const float* b2v = (const float*)d_in[8];
  const float* c1w = (const float*)d_in[9];
  const float* c1b = (const float*)d_in[10];
  const float* c2w = (const float*)d_in[11];
  const float* c2b = (const float*)d_in[12];
  const float* wp  = (const float*)d_in[13];
  const float* l1W = (const float*)d_in[14];
  const float* l1b = (const float*)d_in[15];
  const float* l4W = (const float*)d_in[16];
  const float* l4b = (const float*)d_in[17];
  float* out = (float*)d_out;

  char* ws = (char*)d_ws;
  size_t off = 0;
  auto carve = [&](size_t bytes) -> void* {
    void* p = ws + off;
    off = (off + bytes + 255) & ~(size_t)255;
    return p;
  };
  float* dinv   = (float*)carve(sizeof(float) * N_);       // deg -> dinv in place
  float* hW     = (float*)carve(sizeof(float) * N_ * 32);
  float* agg    = (float*)carve(sizeof(float) * N_ * 32);  // first N floats reused for layer 3
  float* h1     = (float*)carve(sizeof(float) * N_ * 32);
  float* h2     = (float*)carve(sizeof(float) * N_ * 32);
  float* h3     = (float*)carve(sizeof(float) * N_);
  float* hW2    = (float*)carve(sizeof(float) * N_);
  unsigned* hist = (unsigned*)carve(sizeof(unsigned) * NBINS_);
  unsigned* meta = (unsigned*)carve(sizeof(unsigned) * 16);
  unsigned long long* cand = (unsigned long long*)carve(8ull * CAP_);
  int*   idxb   = (int*)carve(sizeof(int) * 1024);
  float* pooled = (float*)carve(sizeof(float) * K_ * LAT_);
  float* z1     = (float*)carve(sizeof(float) * 16 * K_);
  float* feats  = (float*)carve(sizeof(float) * S_ * FEAT_);
  float* inputs = (float*)carve(sizeof(float) * L1IN_);
  float* h128   = (float*)carve(sizeof(float) * 128);

  const int numTiles = N_ / 16;                       // 3125 (exact)
  const int gemmBlocks = (numTiles + 3) / 4;          // 4 waves / block
  const int gN   = (N_ + 255) / 256;
  const int gN32 = (N_ * 32 + 255) / 256;
  const int gE   = (E_ + 255) / 256;
  const int gE8  = E_ / 8;                            // 8 edges / 256-thread block

  for (int s = 0; s < S_; ++s) {
    const float* xs  = x + (size_t)s * N_ * FIN_;
    const int* srcs  = ei + (size_t)s * 2 * E_;
    const int* dsts  = srcs + E_;
    float* featsS    = feats + (size_t)s * FEAT_;

    // degrees (self-loop init) -> dinv
    fill_f32<<<gN, 256, 0, stream>>>(dinv, 1.0f, N_);
    deg_accum<<<gE, 256, 0, stream>>>(dsts, dinv, E_);
    rsqrt_ip<<<gN, 256, 0, stream>>>(dinv, N_);

    // layer 1: hW = x @ W0 ; scatter ; tanh
    gemm_wmma_nx32<FIN_><<<gemmBlocks, 128, 0, stream>>>(xs, W0, hW, numTiles);
    fill_f32<<<gN32, 256, 0, stream>>>(agg, 0.f, N_ * 32);
    scatter32<<<gE8, 256, 0, stream>>>(srcs, dsts, hW, dinv, agg, E_);
    finalize32<<<gN32, 256, 0, stream>>>(agg, hW, dinv, b0, h1, N_ * 32);

    // layer 2: hW = h1 @ W1 ; scatter ; tanh
    gemm_wmma_nx32<32><<<gemmBlocks, 128, 0, stream>>>(h1, W1, hW, numTiles);
    fill_f32<<<gN32, 256, 0, stream>>>(agg, 0.f, N_ * 32);
    scatter32<<<gE8, 256, 0, stream>>>(srcs, dsts, hW, dinv, agg, E_);
    finalize32<<<gN32, 256, 0, stream>>>(agg, hW, dinv, b1, h2, N_ * 32);

    // layer 3: hW2 = h2 @ W2 ; scalar scatter ; tanh
    gemv32<<<gN, 256, 0, stream>>>(h2, W2, hW2, N_);
    fill_f32<<<gN, 256, 0, stream>>>(agg, 0.f, N_);     // reuse front of agg
    scatter1<<<gE, 256, 0, stream>>>(srcs, dsts, hW2, dinv, agg, E_);
    finalize1<<<gN, 256, 0, stream>>>(agg, hW2, dinv, b2v, h3, N_);

    // top-K=600 by last channel (descending, stable index tie-break)
    fill_u32<<<(NBINS_ + 255) / 256, 256, 0, stream>>>(hist, 0u, NBINS_);
    hist_kernel<<<gN, 256, 0, stream>>>(h3, hist, N_);
    find_thresh<<<1, 32, 0, stream>>>(hist, meta);
    compact<<<gN, 256, 0, stream>>>(h3, meta, cand, N_);
    sort_topk<<<1, 1024, 0, stream>>>(cand, meta, idxb);

    // pooling + convs -> per-snapshot feature vector
    emb_kernel<<<1, 96, 0, stream>>>(h1, h2, h3, featsS);
    gather_pool<<<(K_ * LAT_ + 255) / 256, 256, 0, stream>>>(idxb, h1, h2, h3, pooled);
    conv1_kernel<<<(16 * K_ + 255) / 256, 256, 0, stream>>>(pooled, c1w, c1b, z1);
    conv2_kernel<<<(32 * 296 + 255) / 256, 256, 0, stream>>>(z1, c2w, c2b, featsS);
  }

  // temporal combine + dense head
  combine_kernel<<<(L1IN_ + 255) / 256, 256, 0, stream>>>(feats, wp, inputs);
  lin1_kernel<<<128, 256, 0, stream>>>(inputs, l1W, l1b, h128);
  lin4_kernel<<<1, 128, 0, stream>>>(h128, l4W, l4b, out);
}